// ProbProtoSegHead_4827543240789
// MI455X (gfx1250) — compile-verified
//
#include <hip/hip_runtime.h>
#include <math.h>

// ProbProtoSegHead scoring: out[n, c, m] = -0.5 * mean_k[ (pm-x)^2/(xv+pv) + ln(xv+pv) ]
// N=8192, K=64, C*M=190. Transcendental-bound; staged via CDNA5 Tensor Data Mover.

typedef unsigned int u32;
typedef unsigned long long u64;

#define KDIM 64
#define CMTOT 190
#define JCHUNK 95
#define ROWS 8
#define LDS_STRIDE 68   // 64 dwords + 4 pad dwords (TDM pad): bank-conflict-free, 16B aligned

#if defined(__HIP_DEVICE_COMPILE__) && defined(__gfx1250__) && \
    __has_builtin(__builtin_amdgcn_tensor_load_to_lds) && \
    __has_builtin(__builtin_amdgcn_s_wait_tensorcnt)
#define USE_TDM 1
#else
#define USE_TDM 0
#endif

__device__ __forceinline__ float fast_rcp(float s) {
#if defined(__HIP_DEVICE_COMPILE__)
    return __builtin_amdgcn_rcpf(s);      // v_rcp_f32
#else
    return 1.0f / s;
#endif
}

__device__ __forceinline__ float fast_log2(float s) {
#if defined(__HIP_DEVICE_COMPILE__)
    return __builtin_amdgcn_logf(s);      // v_log_f32 (base-2)
#else
    return log2f(s);
#endif
}

#if USE_TDM
typedef __attribute__((ext_vector_type(4))) u32 v4u;
typedef __attribute__((ext_vector_type(8))) int v8i;
typedef __attribute__((ext_vector_type(4))) int v4i;

// Issue one TDM 2D tile load global->LDS per the D# layout (CDNA5 ISA).
// dim0/tile0 in elements (4B each); optional LDS padding of 4 dwords per 64-dword row.
// clang-23 builtin arity is 6: (g0 v4u, g1 v8i, g2 v4i, g3 v4i, v8i, i32 cpol);
// the 5th (uncharacterized) group is passed zero-filled.
__device__ __forceinline__ void tdm_load_2d(u32 lds_byte, const void* g,
                                            u32 dim0, u32 dim1,
                                            u32 tile0, u32 tile1,
                                            u32 stride0, int padded) {
    u64 ga = (u64)(uintptr_t)g;
    v4u g0;
    g0[0] = 1u;                                           // count=1 valid, user mode
    g0[1] = lds_byte;                                     // lds_addr (bytes)
    g0[2] = (u32)ga;                                      // global_addr[31:0]
    g0[3] = (u32)((ga >> 32) & 0x01FFFFFFu) | (2u << 30); // global_addr[56:32] | type=2

    v8i g1;
    u32 w0 = (2u << 16);                                  // data_size = 4 bytes
    if (padded)                                           // pad every 64 dwords by 4 dwords
        w0 |= (1u << 20) | (5u << 22) | (3u << 25);       // pad_enable, interval=64dw, amount=4dw
    g1[0] = (int)w0;
    g1[1] = (int)((dim0 & 0xFFFFu) << 16);                // tensor_dim0[15:0]
    g1[2] = (int)((dim0 >> 16) | ((dim1 & 0xFFFFu) << 16));
    g1[3] = (int)((dim1 >> 16) | ((tile0 & 0xFFFFu) << 16));
    g1[4] = (int)(tile1 & 0xFFFFu);                       // tile_dim1; tile_dim2 = 0
    g1[5] = (int)stride0;                                 // tensor_dim0_stride[31:0]
    g1[6] = 0;
    g1[7] = 0;

    v4i g2; g2[0] = 1; g2[1] = 1; g2[2] = 0; g2[3] = 0;   // tensor_dim2=1, tensor_dim3=1
    v4i g3; g3[0] = 0; g3[1] = 1 << 16; g3[2] = 0; g3[3] = 0; // tensor_dim4=1

    v8i gx;                                               // uncharacterized trailing group
    gx[0] = 0; gx[1] = 0; gx[2] = 0; gx[3] = 0;
    gx[4] = 0; gx[5] = 0; gx[6] = 0; gx[7] = 0;

    __builtin_amdgcn_tensor_load_to_lds(g0, g1, g2, g3, gx, 0);
}
#endif

__global__ __launch_bounds__(256)
void probproto_seg_head_kernel(const float* __restrict__ x,
                               const float* __restrict__ xv,
                               const float* __restrict__ pm,
                               const float* __restrict__ pv,
                               float* __restrict__ out, int N) {
    __shared__ __align__(16) float pm_s[JCHUNK * LDS_STRIDE];
    __shared__ __align__(16) float pv_s[JCHUNK * LDS_STRIDE];
    __shared__ __align__(16) float x_s[ROWS * KDIM];
    __shared__ __align__(16) float xv_s[ROWS * KDIM];

    const int tid = threadIdx.x;
    const int n0 = blockIdx.x * ROWS;
    const int jbase = blockIdx.y * JCHUNK;

#if USE_TDM
    if (tid < 32) {  // wave 0 drives the TDM (EXEC-independent scalar-side op)
        tdm_load_2d((u32)(uintptr_t)&pm_s[0], pm + (size_t)jbase * KDIM,
                    KDIM, JCHUNK, KDIM, JCHUNK, KDIM, 1);
        tdm_load_2d((u32)(uintptr_t)&pv_s[0], pv + (size_t)jbase * KDIM,
                    KDIM, JCHUNK, KDIM, JCHUNK, KDIM, 1);
        tdm_load_2d((u32)(uintptr_t)&x_s[0],  x  + (size_t)n0 * KDIM,
                    KDIM, ROWS, KDIM, ROWS, KDIM, 0);
        tdm_load_2d((u32)(uintptr_t)&xv_s[0], xv + (size_t)n0 * KDIM,
                    KDIM, ROWS, KDIM, ROWS, KDIM, 0);
        __builtin_amdgcn_s_wait_tensorcnt(0);
    }
#else
    for (int i = tid; i < JCHUNK * KDIM; i += 256) {
        int j = i >> 6, k = i & 63;
        pm_s[j * LDS_STRIDE + k] = pm[(size_t)jbase * KDIM + i];
        pv_s[j * LDS_STRIDE + k] = pv[(size_t)jbase * KDIM + i];
    }
    for (int i = tid; i < ROWS * KDIM; i += 256) {
        x_s[i]  = x[(size_t)n0 * KDIM + i];
        xv_s[i] = xv[(size_t)n0 * KDIM + i];
    }
#endif
    __syncthreads();

    const int w = tid >> 5;          // wave id -> row
    const int lane = tid & 31;
    const int n = n0 + w;
    if (n >= N) return;

    const float4* x4  = (const float4*)&x_s[w * KDIM];
    const float4* xv4 = (const float4*)&xv_s[w * KDIM];

    for (int j = lane; j < JCHUNK; j += 32) {
        const float4* pm4 = (const float4*)&pm_s[j * LDS_STRIDE];
        const float4* pv4 = (const float4*)&pv_s[j * LDS_STRIDE];
        float acc = 0.0f;    // sum of d^2 * rcp(s)
        float lacc = 0.0f;   // sum of log2 over 16-element products
#pragma unroll
        for (int c = 0; c < 4; ++c) {          // 4 chunks of 16 k-values
            float prod = 1.0f;
#pragma unroll
            for (int t = 0; t < 4; ++t) {      // 4x float4 per chunk (ds_load_b128)
                const int k4 = c * 4 + t;
                float4 P = pm4[k4], Q = pv4[k4];
                float4 A = x4[k4],  V = xv4[k4];
                float d, s;
                d = P.x - A.x; s = V.x + Q.x; acc = __builtin_fmaf(d * d, fast_rcp(s), acc); prod *= s;
                d = P.y - A.y; s = V.y + Q.y; acc = __builtin_fmaf(d * d, fast_rcp(s), acc); prod *= s;
                d = P.z - A.z; s = V.z + Q.z; acc = __builtin_fmaf(d * d, fast_rcp(s), acc); prod *= s;
                d = P.w - A.w; s = V.w + Q.w; acc = __builtin_fmaf(d * d, fast_rcp(s), acc); prod *= s;
            }
            lacc += fast_log2(prod);           // one v_log_f32 per 16 elements
        }
        // -0.5 * mean_k;  ln(s) = ln2 * log2(s)
        float simv = -0.0078125f * __builtin_fmaf(0.69314718055994531f, lacc, acc);
        out[(size_t)n * CMTOT + jbase + j] = simv;
    }
}

extern "C" void kernel_launch(void* const* d_in, const int* in_sizes, int n_in,
                              void* d_out, int out_size, void* d_ws, size_t ws_size,
                              hipStream_t stream) {
    const float* x  = (const float*)d_in[0];
    const float* xv = (const float*)d_in[1];
    const float* pm = (const float*)d_in[2];
    const float* pv = (const float*)d_in[3];
    float* out = (float*)d_out;
    const int N = in_sizes[0] / KDIM;  // 8192
    dim3 grid((unsigned)((N + ROWS - 1) / ROWS),
              (unsigned)((CMTOT + JCHUNK - 1) / JCHUNK));
    probproto_seg_head_kernel<<<grid, 256, 0, stream>>>(x, xv, pm, pv, out, N);
}